// USMSharp_24137716203640
// MI455X (gfx1250) — compile-verified
//
#include <hip/hip_runtime.h>
#include <math.h>

#define TW 512
#define THH 512
#define NPIX (TW * THH)
#define PLANES 48          // 16 * 3
#define RAD 25
#define KTAPS 51
#define CHUNKS 17          // ceil((16 + 2*RAD)/4) = ceil(66/4)
#define WAVES_PER_BLOCK 8
#define TILES_PER_WAVE 8   // 8 consecutive column tiles per wave

typedef float v2f __attribute__((ext_vector_type(2)));
typedef float v8f __attribute__((ext_vector_type(8)));
typedef v2f v2f_u __attribute__((aligned(4)));   // possibly 4B-aligned pair load

struct GParams { float g[KTAPS]; };

__device__ __forceinline__ int clampi(int v, int lo, int hi) {
    return v < lo ? lo : (v > hi ? hi : v);
}

__device__ __forceinline__ float tap(const float* sg, int i) {
    float v = sg[clampi(i, 0, KTAPS - 1)];
    return (i >= 0 && i < KTAPS) ? v : 0.0f;
}

__device__ __forceinline__ v8f wmma_acc(v2f a, v2f b, v8f c) {
    // D = A(16x4) * B(4x16) + C, fp32 in / fp32 acc
    return __builtin_amdgcn_wmma_f32_16x16x4_f32(
        false, a, false, b, (short)0, c, false, false);
}

__device__ __forceinline__ v2f load2(const float* p) { return *(const v2f_u*)p; }

// Precompute the 17 Toeplitz fragments for this lane (band g[k - n], k=4j+2*lhi+{0,1}).
__device__ __forceinline__ void load_taps(const float* sg, int lane, v2f* tb) {
    const int lhi = lane >> 4, llo = lane & 15;
#pragma unroll
    for (int j = 0; j < CHUNKS; ++j) {
        const int kb = 4 * j + 2 * lhi;
        tb[j] = (v2f){tap(sg, kb - llo), tap(sg, kb + 1 - llo)};
    }
}

// wave id -> (plane, row-tile, column-tile-group); 128 waves per plane
__device__ __forceinline__ void decode(int& plane, int& m0, int& g0, int& lane) {
    const int tid = threadIdx.x;
    lane = tid & 31;
    const int wg = blockIdx.x * WAVES_PER_BLOCK + (tid >> 5);
    plane = wg >> 7;
    const int rem = wg & 127;
    m0 = (rem >> 2) << 4;          // row-tile origin
    g0 = (rem & 3) * (16 * TILES_PER_WAVE);  // first column of the 8-tile group
}

// ================= Pass 1 & 3: horizontal 51-tap blur =================
__global__ __launch_bounds__(256) void usm_hpass(const float* __restrict__ in,
                                                 float* __restrict__ out, GParams P) {
    __shared__ float sg[KTAPS];
    if (threadIdx.x < KTAPS) sg[threadIdx.x] = P.g[threadIdx.x];
    __syncthreads();
    int plane, m0, g0, lane;
    decode(plane, m0, g0, lane);
    const int lhi = lane >> 4, llo = lane & 15;

    v2f tb[CHUNKS];
    load_taps(sg, lane, tb);       // B = Toeplitz, reused for all 8 tiles

    const float* rowp = in + plane * NPIX + (m0 + llo) * TW;         // A row for this lane
    const float* ap   = rowp + g0 - RAD + 2 * lhi;                   // chunk-0 base, tile 0
    float*       op   = out + plane * NPIX + (m0 + 8 * lhi) * TW + llo + g0;

    int n0 = g0;
    for (int t = 0; t < TILES_PER_WAVE; ++t, n0 += 16, ap += 16, op += 16) {
        v8f c = {0.f, 0.f, 0.f, 0.f, 0.f, 0.f, 0.f, 0.f};
        const int interior =
            __builtin_amdgcn_readfirstlane((n0 >= RAD) & (n0 + 42 < TW));
        if (interior) {
#pragma unroll
            for (int j = 0; j < CHUNKS; ++j)           // b64 load at imm offset 16*j
                c = wmma_acc(load2(ap + 4 * j), tb[j], c);
        } else {
#pragma unroll
            for (int j = 0; j < CHUNKS; ++j) {
                const int c0 = n0 - RAD + 2 * lhi + 4 * j;
                float a0 = rowp[clampi(c0, 0, TW - 1)];
                float a1 = rowp[clampi(c0 + 1, 0, TW - 1)];
                a0 = (c0 >= 0 && c0 < TW) ? a0 : 0.0f;
                a1 = (c0 + 1 >= 0 && c0 + 1 < TW) ? a1 : 0.0f;
                c = wmma_acc((v2f){a0, a1}, tb[j], c);
            }
        }
#pragma unroll
        for (int i = 0; i < 8; ++i) op[i * TW] = c[i];
    }
}

// ================= Vertical conv core (Toeplitz as A, image tile as B) =================
// Returns blur tile accumulator for tile with top row m0, columns [n0, n0+16).
__device__ __forceinline__ v8f vconv_interior(const float* bp, const v2f* ta) {
    v8f c = {0.f, 0.f, 0.f, 0.f, 0.f, 0.f, 0.f, 0.f};
#pragma unroll
    for (int j = 0; j < CHUNKS; ++j) {                 // b32 loads at imm offsets j*8192(+2048)
        v2f b = {bp[j * 4 * TW], bp[j * 4 * TW + TW]};
        c = wmma_acc(ta[j], b, c);
    }
    return c;
}

__device__ __forceinline__ v8f vconv_guarded(const float* colp, const v2f* ta,
                                             int m0, int lhi) {
    v8f c = {0.f, 0.f, 0.f, 0.f, 0.f, 0.f, 0.f, 0.f};
#pragma unroll
    for (int j = 0; j < CHUNKS; ++j) {
        const int r0 = m0 - RAD + 2 * lhi + 4 * j;
        float b0 = colp[clampi(r0, 0, THH - 1) * TW];
        float b1 = colp[clampi(r0 + 1, 0, THH - 1) * TW];
        b0 = (r0 >= 0 && r0 < THH) ? b0 : 0.0f;
        b1 = (r0 + 1 >= 0 && r0 + 1 < THH) ? b1 : 0.0f;
        c = wmma_acc(ta[j], (v2f){b0, b1}, c);
    }
    return c;
}

// ================= Pass 2: vertical blur -> mask + sharp =================
__global__ __launch_bounds__(256) void usm_vpass_mask(const float* __restrict__ tmp,
                                                      const float* __restrict__ img,
                                                      float* __restrict__ mask,
                                                      float* __restrict__ sharp, GParams P) {
    __shared__ float sg[KTAPS];
    if (threadIdx.x < KTAPS) sg[threadIdx.x] = P.g[threadIdx.x];
    __syncthreads();
    int plane, m0, g0, lane;
    decode(plane, m0, g0, lane);
    const int lhi = lane >> 4, llo = lane & 15;

    v2f ta[CHUNKS];
    load_taps(sg, lane, ta);       // A = Toeplitz

    const int interior =
        __builtin_amdgcn_readfirstlane((m0 >= RAD) & (m0 + 42 < THH));
    const float* bp   = tmp + plane * NPIX + (m0 - RAD + 2 * lhi) * TW + llo + g0;
    const float* colp = tmp + plane * NPIX + llo + g0;
    const int    obase0 = plane * NPIX + (m0 + 8 * lhi) * TW + llo + g0;

    int obase = obase0;
    for (int t = 0; t < TILES_PER_WAVE; ++t, bp += 16, colp += 16, obase += 16) {
        v8f c = interior ? vconv_interior(bp, ta)
                         : vconv_guarded(colp, ta, m0, lhi);   // c = blur tile
#pragma unroll
        for (int i = 0; i < 8; ++i) {
            const int idx = obase + i * TW;
            const float iv  = img[idx];
            const float res = iv - c[i];
            mask[idx] = (fabsf(res) * 255.0f > 10.0f) ? 1.0f : 0.0f;
            sharp[idx] = fminf(fmaxf(iv + 0.5f * res, 0.0f), 1.0f);
        }
    }
}

// ================= Pass 4: vertical blur of mask -> final blend =================
__global__ __launch_bounds__(256) void usm_vpass_final(const float* __restrict__ tmp,
                                                       const float* __restrict__ img,
                                                       const float* __restrict__ sharp,
                                                       float* __restrict__ out, GParams P) {
    __shared__ float sg[KTAPS];
    if (threadIdx.x < KTAPS) sg[threadIdx.x] = P.g[threadIdx.x];
    __syncthreads();
    int plane, m0, g0, lane;
    decode(plane, m0, g0, lane);
    const int lhi = lane >> 4, llo = lane & 15;

    v2f ta[CHUNKS];
    load_taps(sg, lane, ta);

    const int interior =
        __builtin_amdgcn_readfirstlane((m0 >= RAD) & (m0 + 42 < THH));
    const float* bp   = tmp + plane * NPIX + (m0 - RAD + 2 * lhi) * TW + llo + g0;
    const float* colp = tmp + plane * NPIX + llo + g0;
    int obase = plane * NPIX + (m0 + 8 * lhi) * TW + llo + g0;

    for (int t = 0; t < TILES_PER_WAVE; ++t, bp += 16, colp += 16, obase += 16) {
        v8f c = interior ? vconv_interior(bp, ta)
                         : vconv_guarded(colp, ta, m0, lhi);   // c = soft_mask tile
#pragma unroll
        for (int i = 0; i < 8; ++i) {
            const int idx = obase + i * TW;
            const float iv = img[idx];
            const float sm = c[i];
            out[idx] = sm * sharp[idx] + (1.0f - sm) * iv;
        }
    }
}

extern "C" void kernel_launch(void* const* d_in, const int* in_sizes, int n_in,
                              void* d_out, int out_size, void* d_ws, size_t ws_size,
                              hipStream_t stream) {
    (void)in_sizes; (void)n_in; (void)out_size; (void)ws_size;
    const float* img = (const float*)d_in[0];
    // d_in[1] is the 51x51 outer-product kernel; rebuild the exact 1-D factor on the
    // host in float64 (matches numpy: sigma = 0.3*((51-1)*0.5-1)+0.8 = 8.0).
    float* out = (float*)d_out;
    float* ws0 = (float*)d_ws;                       // 48-plane temp (H-pass output)
    float* ws1 = ws0 + (size_t)PLANES * NPIX;        // sharp image

    GParams P;
    {
        const double sigma = 0.3 * ((KTAPS - 1) * 0.5 - 1.0) + 0.8;
        double raw[KTAPS], s = 0.0;
        for (int i = 0; i < KTAPS; ++i) {
            const double x = i - (KTAPS - 1) / 2.0;
            raw[i] = exp(-(x * x) / (2.0 * sigma * sigma));
            s += raw[i];
        }
        for (int i = 0; i < KTAPS; ++i) P.g[i] = (float)(raw[i] / s);
    }

    const dim3 blk(256);
    // 48 planes * 32 row-tiles * 4 column-groups = 6144 waves = 768 blocks
    const dim3 grd(PLANES * 32 * 4 / WAVES_PER_BLOCK);

    usm_hpass<<<grd, blk, 0, stream>>>(img, ws0, P);                 // tmp  = Hblur(img)
    usm_vpass_mask<<<grd, blk, 0, stream>>>(ws0, img, out, ws1, P);  // mask->out, sharp->ws1
    usm_hpass<<<grd, blk, 0, stream>>>(out, ws0, P);                 // tmp  = Hblur(mask)
    usm_vpass_final<<<grd, blk, 0, stream>>>(ws0, img, ws1, out, P); // blend -> out
}